// RepBEV_vit_47648367182437
// MI455X (gfx1250) — compile-verified
//
#include <hip/hip_runtime.h>
#include <hip/hip_bf16.h>
#include <stdint.h>

// ---------------- problem constants (from reference) ----------------
#define FWv 22
#define FHv 8
#define Dv  41
#define Nv  6
#define Bv  8
#define Cv  64
#define NPTS (Bv*Nv*Dv*FHv*FWv)          // 346368 points
#define NXv 200
#define NYv 200
#define ACC_ELEMS ((size_t)Bv*NXv*NYv*Cv) // 20,480,000 floats (81.92 MB)
#define TP  68                            // padded LDS row stride (16B aligned, 2-way bank conflict)

// ---------------- CDNA5-specific inline-asm helpers ----------------
// Hardware f32 atomic add at device scope (atomics RMW at L2; output fits in 192MB L2).
__device__ __forceinline__ void gatomic_add_f32(float* p, float v) {
  asm volatile("global_atomic_add_f32 %0, %1, off scope:SCOPE_DEV"
               :: "v"((uint64_t)(uintptr_t)p), "v"(v)
               : "memory");
}

// ---------------- small matrix helpers ----------------
__device__ __forceinline__ void inv3(const float a[9], float o[9]) {
  float c00 =  a[4]*a[8] - a[5]*a[7];
  float c01 = -(a[3]*a[8] - a[5]*a[6]);
  float c02 =  a[3]*a[7] - a[4]*a[6];
  float det =  a[0]*c00 + a[1]*c01 + a[2]*c02;
  float id  = 1.0f / det;
  o[0] =  c00*id;
  o[1] = -(a[1]*a[8]-a[2]*a[7])*id;
  o[2] =  (a[1]*a[5]-a[2]*a[4])*id;
  o[3] =  c01*id;
  o[4] =  (a[0]*a[8]-a[2]*a[6])*id;
  o[5] = -(a[0]*a[5]-a[2]*a[3])*id;
  o[6] =  c02*id;
  o[7] = -(a[0]*a[7]-a[1]*a[6])*id;
  o[8] =  (a[0]*a[4]-a[1]*a[3])*id;
}

__device__ __forceinline__ void mm3(const float a[9], const float b[9], float o[9]) {
#pragma unroll
  for (int i = 0; i < 3; ++i)
#pragma unroll
    for (int j = 0; j < 3; ++j)
      o[i*3+j] = a[i*3+0]*b[0*3+j] + a[i*3+1]*b[1*3+j] + a[i*3+2]*b[2*3+j];
}

// ---------------- kernels ----------------
__global__ void __launch_bounds__(256) bev_zero(float4* __restrict__ p, int n4) {
  int i = blockIdx.x * 256 + threadIdx.x;
  if (i < n4) p[i] = make_float4(0.f, 0.f, 0.f, 0.f);
}

// One wave32 per frustum point; lane handles channels (2*lane, 2*lane+1).
// direct==0: accumulate into channel-contiguous scratch (B,X,Y,C)  -> contiguous atomics
// direct==1: accumulate straight into d_out (B,C,X,Y)              -> fallback
__global__ void __launch_bounds__(256) bev_scatter(
    const float* __restrict__ x,
    const float* __restrict__ rots,
    const float* __restrict__ trans,
    const float* __restrict__ intrins,
    const float* __restrict__ post_rots,
    const float* __restrict__ post_trans,
    float* __restrict__ dst, int direct)
{
  int gtid = blockIdx.x * 256 + threadIdx.x;
  int pnt  = gtid >> 5;          // wave-uniform point id
  int lane = threadIdx.x & 31;
  if (pnt >= NPTS) return;

  // decompose point index: (B, N, D, FH, FW)
  int t  = pnt;
  int w  = t % FWv; t /= FWv;
  int h  = t % FHv; t /= FHv;
  int di = t % Dv;  t /= Dv;
  int n  = t % Nv;
  int b  = t / Nv;
  int bn = b * Nv + n;

  const float* R  = rots       + bn*9;
  const float* T  = trans      + bn*3;
  const float* Kc = intrins    + bn*9;
  const float* P  = post_rots  + bn*9;
  const float* pt = post_trans + bn*3;

  // frustum pixel coords (linspace endpoints inclusive)
  float u   = (float)w * (351.0f / 21.0f);
  float v   = (float)h * (127.0f / 7.0f);
  float dep = 4.0f + (float)di;

  float Pl[9], Kl[9], iP[9], iK[9], M[9];
#pragma unroll
  for (int i = 0; i < 9; ++i) { Pl[i] = P[i]; Kl[i] = Kc[i]; }
  inv3(Pl, iP);
  inv3(Kl, iK);
  float Rl[9];
#pragma unroll
  for (int i = 0; i < 9; ++i) Rl[i] = R[i];
  mm3(Rl, iK, M);   // combine = rots @ inv(intrins)

  // geometry: undo post-aug, un-project, cam->ego
  float q0 = u   - pt[0];
  float q1 = v   - pt[1];
  float q2 = dep - pt[2];
  float r0 = iP[0]*q0 + iP[1]*q1 + iP[2]*q2;
  float r1 = iP[3]*q0 + iP[4]*q1 + iP[5]*q2;
  float r2 = iP[6]*q0 + iP[7]*q1 + iP[8]*q2;
  float s0 = r0 * r2, s1 = r1 * r2, s2 = r2;
  float gxf = M[0]*s0 + M[1]*s1 + M[2]*s2 + T[0];
  float gyf = M[3]*s0 + M[4]*s1 + M[5]*s2 + T[1];
  float gzf = M[6]*s0 + M[7]*s1 + M[8]*s2 + T[2];

  // voxelize: (geom - (bx - dx/2)) / dx, truncation-toward-zero == astype(int32)
  int g0 = (int)((gxf + 50.0f) * 2.0f);
  int g1 = (int)((gyf + 50.0f) * 2.0f);
  int g2 = (int)((gzf + 10.0f) / 20.0f);
  if (g0 < 0 || g0 >= NXv || g1 < 0 || g1 >= NYv || g2 != 0) return;

  // coalesced float2 feature load (64 contiguous floats per point)
  const float2 xv = *(const float2*)(x + (size_t)pnt * Cv + lane * 2);
  int c0 = lane * 2;

  if (direct) {
    // (B, C, X, Y): channel stride 40000
    size_t o = (((size_t)(b*Cv + c0) * NXv + g0) * NYv) + g1;
    gatomic_add_f32(dst + o,         xv.x);
    gatomic_add_f32(dst + o + 40000, xv.y);
  } else {
    // (B, X, Y, C): one contiguous 256B run per point
    size_t o = ((((size_t)b*NXv + g0) * NYv + g1) * Cv) + c0;
    gatomic_add_f32(dst + o,     xv.x);
    gatomic_add_f32(dst + o + 1, xv.y);
  }
}

// Transpose (B,X,Y,C) scratch -> (B,C,X,Y) output, one block per (b, x-slice).
// Stages the 200x64 tile through LDS via CDNA5 async global->LDS copies.
__global__ void __launch_bounds__(256) bev_transpose(const float* __restrict__ acc,
                                                     float* __restrict__ out)
{
  __shared__ float tile[NYv * TP];   // 200*68*4 = 54,400 B
  int blk = blockIdx.x;              // 0..1599
  int b   = blk / NXv;
  int gx  = blk % NXv;

  const float* src = acc + (((size_t)b * NXv + gx) * NYv) * Cv;  // 12,800 contiguous floats
  unsigned ldsBase = (unsigned)(uintptr_t)(void*)tile;           // low 32 bits = LDS offset

  // async-copy 200 rows of 64 floats into padded LDS rows (16 x b128 per row)
  for (int i = threadIdx.x; i < NYv * (Cv / 4); i += 256) {      // 3200 float4 transfers
    int y = i >> 4;
    int j = (i & 15) << 2;
    unsigned ldsAddr = ldsBase + (unsigned)((y * TP + j) * 4);
    uint64_t ga = (uint64_t)(uintptr_t)(src + (size_t)y * Cv + j);
    asm volatile("global_load_async_to_lds_b128 %0, %1, off"
                 :: "v"(ldsAddr), "v"(ga)
                 : "memory");
  }
  asm volatile("s_wait_asynccnt 0" ::: "memory");
  __syncthreads();

  // write out[b, c, gx, :] — 200-float contiguous runs, fully coalesced
  float* dstb = out + (size_t)b * Cv * NXv * NYv + (size_t)gx * NYv;
  for (int o = threadIdx.x; o < Cv * NYv; o += 256) {            // 12,800 floats
    int c = o / NYv;
    int y = o - c * NYv;
    dstb[(size_t)c * NXv * NYv + y] = tile[y * TP + c];
  }
}

// ---------------- host launcher ----------------
extern "C" void kernel_launch(void* const* d_in, const int* in_sizes, int n_in,
                              void* d_out, int out_size, void* d_ws, size_t ws_size,
                              hipStream_t stream) {
  const float* x          = (const float*)d_in[0];
  const float* rots       = (const float*)d_in[1];
  const float* trans      = (const float*)d_in[2];
  const float* intrins    = (const float*)d_in[3];
  const float* post_rots  = (const float*)d_in[4];
  const float* post_trans = (const float*)d_in[5];
  float* out = (float*)d_out;

  const size_t accBytes = ACC_ELEMS * sizeof(float);
  const int n4 = (int)(ACC_ELEMS / 4);                 // 5,120,000 float4
  const int zeroBlocks = (n4 + 255) / 256;             // 20,000
  const int scatBlocks = (NPTS * 32 + 255) / 256;      // 43,296

  const bool useWs = (d_ws != nullptr) && (ws_size >= accBytes);

  if (useWs) {
    float* acc = (float*)d_ws;
    bev_zero<<<zeroBlocks, 256, 0, stream>>>((float4*)acc, n4);
    bev_scatter<<<scatBlocks, 256, 0, stream>>>(x, rots, trans, intrins,
                                                post_rots, post_trans, acc, 0);
    bev_transpose<<<Bv * NXv, 256, 0, stream>>>(acc, out);
  } else {
    bev_zero<<<zeroBlocks, 256, 0, stream>>>((float4*)out, n4);
    bev_scatter<<<scatBlocks, 256, 0, stream>>>(x, rots, trans, intrins,
                                                post_rots, post_trans, out, 1);
  }
}